// R_Binarization_48215302864976
// MI455X (gfx1250) — compile-verified
//
#include <hip/hip_runtime.h>
#include <stdint.h>

#define HW   (4096 * 4096)        // pixels per plane (2^24)
#define NTOT (3 * HW)             // total input floats
#define NBLK 2048
#define NTHR 256

typedef __attribute__((ext_vector_type(2))) float v2f;
typedef __attribute__((ext_vector_type(4))) float vf4;
typedef __attribute__((ext_vector_type(8))) float v8f;

// ---------------- init: zero ws (hist + max + threshold) -------------------
__global__ void k_init(uint32_t* hist, uint32_t* maxbits, int* thresh) {
    hist[threadIdx.x] = 0u;
    if (threadIdx.x == 0) { *maxbits = 0u; *thresh = 0; }
}

// ---------------- pass 1: global max (positive floats -> uint order) -------
__global__ void k_max(const vf4* __restrict__ x4, int n4, uint32_t* __restrict__ maxbits) {
    int stride = gridDim.x * blockDim.x;
    float m = 0.0f;
    for (int i = blockIdx.x * blockDim.x + threadIdx.x; i < n4; i += stride) {
        vf4 v = x4[i];
        m = fmaxf(m, fmaxf(fmaxf(v.x, v.y), fmaxf(v.z, v.w)));
    }
    for (int off = 16; off; off >>= 1) m = fmaxf(m, __shfl_xor(m, off, 32));
    __shared__ float smax[8];
    int lane = threadIdx.x & 31, wid = threadIdx.x >> 5;
    if (lane == 0) smax[wid] = m;
    __syncthreads();
    if (wid == 0) {
        m = (lane < (int)(blockDim.x >> 5)) ? smax[lane] : 0.0f;
        for (int off = 16; off; off >>= 1) m = fmaxf(m, __shfl_xor(m, off, 32));
        if (lane == 0) atomicMax(maxbits, __float_as_uint(m)); // positive floats: uint order == float order
    }
}

// ---------------- pass 2: quantize + gray + histogram ----------------------
__device__ __forceinline__ float quant_u8(float v, float mx) {
    // reference: floor(v * 255.0 / max), clip [0,255]; keep exact op order (no FMA)
    float s = __fdiv_rn(__fmul_rn(v, 255.0f), mx);
    return fminf(fmaxf(floorf(s), 0.0f), 255.0f);
}
__device__ __forceinline__ float gray_of(float r, float g, float b, float mx) {
    float R = quant_u8(r, mx), G = quant_u8(g, mx), B = quant_u8(b, mx);
    // (wr*R + wg*G) + wb*B, round-to-nearest-even, no contraction
    float s = __fadd_rn(__fadd_rn(__fmul_rn(0.299f, R), __fmul_rn(0.587f, G)),
                        __fmul_rn(0.114f, B));
    return fminf(fmaxf(rintf(s), 0.0f), 255.0f);
}

__global__ void k_gray_hist(const float* __restrict__ x, float* __restrict__ gray_out,
                            const uint32_t* __restrict__ maxbits, uint32_t* __restrict__ ghist) {
    __shared__ uint32_t lh[256];
    for (int i = threadIdx.x; i < 256; i += blockDim.x) lh[i] = 0u;
    __syncthreads();
    const float mx = __uint_as_float(*maxbits);
    const vf4* r4 = (const vf4*)x;
    const vf4* g4 = (const vf4*)(x + HW);
    const vf4* b4 = (const vf4*)(x + 2 * HW);
    vf4* o4 = (vf4*)gray_out;
    const int n4 = HW / 4;
    int stride = gridDim.x * blockDim.x;
    for (int i = blockIdx.x * blockDim.x + threadIdx.x; i < n4; i += stride) {
        vf4 r = r4[i], g = g4[i], b = b4[i];
        vf4 o;
        o.x = gray_of(r.x, g.x, b.x, mx);
        o.y = gray_of(r.y, g.y, b.y, mx);
        o.z = gray_of(r.z, g.z, b.z, mx);
        o.w = gray_of(r.w, g.w, b.w, mx);
        atomicAdd(&lh[(int)o.x], 1u);   // ds_add_u32
        atomicAdd(&lh[(int)o.y], 1u);
        atomicAdd(&lh[(int)o.z], 1u);
        atomicAdd(&lh[(int)o.w], 1u);
        o4[i] = o;                       // stash gray in d_out plane 0
    }
    __syncthreads();
    for (int i = threadIdx.x; i < 256; i += blockDim.x)
        if (lh[i]) atomicAdd(&ghist[i], lh[i]);
}

// ---------------- pass 3: Otsu, single wave32, cumsum via WMMA -------------
// Inclusive prefix-sum of 256 values as a triangular matmul:
//   P (16x16, row r = elems 16r..16r+15) times U (upper-triangular ones)
//   => S[r][n] = sum_{k<=n} P[r][k]; done as 4 chained V_WMMA_F32_16X16X4_F32
//   (K-slices of U), then a 16-wide cross-row exclusive scan of row totals.
template <class F>
__device__ __forceinline__ void cumsum256_wmma(F w, float out[8], int lane) {
    const int l = lane & 15, half = lane >> 4;
    v8f acc = {};
#pragma unroll
    for (int c = 0; c < 4; ++c) {
        const int k0 = 4 * c + 2 * half;
        v2f a, b;
        // A 16x4 layout: lanes 0-15 hold (M=l, K=k0/k0+1); lanes 16-31 K=k0+2/k0+3
        a.x = w(16 * l + k0);
        a.y = w(16 * l + k0 + 1);
        // B 4x16 = U rows 4c..4c+3: U[k][n] = (k <= n)
        b.x = (k0 <= l) ? 1.0f : 0.0f;
        b.y = (k0 + 1 <= l) ? 1.0f : 0.0f;
        acc = __builtin_amdgcn_wmma_f32_16x16x4_f32(false, a, false, b,
                                                    (short)0, acc, false, false);
    }
    // row totals: total[r] = S[r][15]; C/D layout puts M=j in lanes 0-15, M=j+8 in 16-31
    float tot[16];
#pragma unroll
    for (int j = 0; j < 8; ++j) {
        tot[j]     = __shfl(acc[j], 15, 32);
        tot[j + 8] = __shfl(acc[j], 31, 32);
    }
    float pre[16];
    pre[0] = 0.0f;
#pragma unroll
    for (int r = 1; r < 16; ++r) pre[r] = pre[r - 1] + tot[r - 1];
#pragma unroll
    for (int j = 0; j < 8; ++j) out[j] = acc[j] + pre[j + 8 * half];
    // lane holds cumsum[16*(j + 8*half) + l] in out[j]
}

__global__ __launch_bounds__(32) void k_otsu(const uint32_t* __restrict__ hist,
                                             int* __restrict__ thresh) {
    const int lane = threadIdx.x;
    const float invn = 1.0f / (float)HW;  // 2^-24, exact
    auto w1 = [&](int i) { return (float)hist[i] * invn; };                    // p
    auto w2 = [&](int i) { return (float)i * ((float)hist[i] * invn); };       // bins*p
    float O[8], M[8];
    cumsum256_wmma(w1, O, lane);   // omega
    cumsum256_wmma(w2, M, lane);   // mu
    const float mu_t = __shfl(M[7], 31, 32);  // mu[255]
    const int l = lane & 15, half = lane >> 4;
    float best = -1.0f;
    int bidx = 0;
#pragma unroll
    for (int j = 0; j < 8; ++j) {
        int t = 16 * (j + 8 * half) + l;
        float om = O[j], mu = M[j];
        float denom = om * (1.0f - om);
        float s = 0.0f;
        if (denom > 0.0f) { float d = mu_t * om - mu; s = d * d / denom; }
        if (s > best) { best = s; bidx = t; }  // ascending t per lane -> first max kept
    }
    for (int off = 16; off; off >>= 1) {       // argmax, first-index tie-break
        float ov = __shfl_xor(best, off, 32);
        int   oi = __shfl_xor(bidx, off, 32);
        if (ov > best || (ov == best && oi < bidx)) { best = ov; bidx = oi; }
    }
    if (lane == 0) *thresh = bidx;
}

// ---------------- pass 4: binarize + GRAY2RGB ------------------------------
__global__ void k_binarize(float* __restrict__ out, const int* __restrict__ thresh) {
    const float tf = (float)(*thresh);   // uniform -> scalar load
    vf4* o0 = (vf4*)out;
    vf4* o1 = (vf4*)(out + HW);
    vf4* o2 = (vf4*)(out + 2 * HW);
    const int n4 = HW / 4;
    int stride = gridDim.x * blockDim.x;
    for (int i = blockIdx.x * blockDim.x + threadIdx.x; i < n4; i += stride) {
        vf4 g = __builtin_nontemporal_load(&o0[i]);   // last use of gray
        vf4 b;
        b.x = g.x > tf ? 1.0f : 0.0f;
        b.y = g.y > tf ? 1.0f : 0.0f;
        b.z = g.z > tf ? 1.0f : 0.0f;
        b.w = g.w > tf ? 1.0f : 0.0f;
        __builtin_nontemporal_store(b, &o0[i]);       // streaming output
        __builtin_nontemporal_store(b, &o1[i]);
        __builtin_nontemporal_store(b, &o2[i]);
    }
}

extern "C" void kernel_launch(void* const* d_in, const int* in_sizes, int n_in,
                              void* d_out, int out_size, void* d_ws, size_t ws_size,
                              hipStream_t stream) {
    (void)in_sizes; (void)n_in; (void)out_size; (void)ws_size;
    const float* x = (const float*)d_in[0];
    float* out = (float*)d_out;
    uint32_t* hist   = (uint32_t*)d_ws;                       // 256 * 4B
    uint32_t* maxbit = (uint32_t*)((char*)d_ws + 1024);       // 4B
    int*      thresh = (int*)((char*)d_ws + 1028);            // 4B

    k_init<<<1, 256, 0, stream>>>(hist, maxbit, thresh);
    k_max<<<NBLK, NTHR, 0, stream>>>((const vf4*)x, NTOT / 4, maxbit);
    k_gray_hist<<<NBLK, NTHR, 0, stream>>>(x, out, maxbit, hist);
    k_otsu<<<1, 32, 0, stream>>>(hist, thresh);
    k_binarize<<<NBLK, NTHR, 0, stream>>>(out, thresh);
}